// SynapticSNN_4870492914051
// MI455X (gfx1250) — compile-verified
//
#include <hip/hip_runtime.h>

typedef __attribute__((ext_vector_type(2))) float v2f;
typedef __attribute__((ext_vector_type(8))) float v8f;

#define NIN  256
#define NH   1024
#define ROWS_PER_BLOCK 16
#define SNN_ALPHA 0.9f
#define NUM_STEPS 50

// Fully fused: x@W1^T+b1 (fp32 WMMA) -> 50-step synaptic recurrence in
// registers -> per-column scale by W2 -> LDS row reduction -> out[B] (+b2).
__global__ __launch_bounds__(256)
void snn_fused_kernel(const float* __restrict__ x,
                      const float* __restrict__ W1,
                      const float* __restrict__ b1,
                      const float* __restrict__ thr,
                      const float* __restrict__ beta_p,
                      const float* __restrict__ W2,
                      const float* __restrict__ b2,
                      float* __restrict__ out)
{
    __shared__ float xs[ROWS_PER_BLOCK * NIN];   // 16 KB x-tile, row-major
    __shared__ float rowsum[ROWS_PER_BLOCK];

    const int tid = threadIdx.x;
    const int m0  = blockIdx.x * ROWS_PER_BLOCK;

    // Cooperative, coalesced load of the x tile [16 x 256] as float4.
    {
        const float4* src = (const float4*)(x + (size_t)m0 * NIN);
        float4*       dst = (float4*)xs;
        #pragma unroll
        for (int i = 0; i < (ROWS_PER_BLOCK * NIN / 4) / 256; ++i)
            dst[tid + i * 256] = src[tid + i * 256];
    }
    if (tid < ROWS_PER_BLOCK) rowsum[tid] = 0.0f;
    __syncthreads();

    const int wave   = tid >> 5;
    const int lane   = tid & 31;
    const int laneN  = lane & 15;   // N column within 16x16 tile (also M-row source for A)
    const int laneHi = lane >> 4;   // lane half selects K pair {0,1} vs {2,3}

    float beta_c = beta_p[0];
    beta_c = fminf(fmaxf(beta_c, 0.0f), 1.0f);   // jnp.clip(beta, 0, 1)

    float acc[8];
    #pragma unroll
    for (int i = 0; i < 8; ++i) acc[i] = 0.0f;

    // A operand source: x-tile row = laneN, pre-offset by 2*laneHi for K-half.
    const float* xrow = xs + laneN * NIN + 2 * laneHi;

    // 8 tiles/wave * 8 waves = 64 N-tiles = NH/16.
    for (int t = 0; t < (NH / 16) / 8; ++t) {
        const int n0 = wave * 128 + t * 16;
        const int n  = n0 + laneN;                 // this lane's output column
        const float* wrow = W1 + (size_t)n * NIN + 2 * laneHi;

        // ---- GEMM tile: K=256 as 64 x V_WMMA_F32_16X16X4_F32 ----
        v8f c = {};
        #pragma unroll 8
        for (int kk = 0; kk < NIN / 4; ++kk) {
            v2f a = *(const v2f*)(xrow + kk * 4);  // ds_load_b64
            v2f b = *(const v2f*)(wrow + kk * 4);  // global_load_b64 (L2 resident)
            c = __builtin_amdgcn_wmma_f32_16x16x4_f32(
                    /*neg_a=*/false, a, /*neg_b=*/false, b,
                    /*c_mod=*/(short)0, c, /*reuse_a=*/false, /*reuse_b=*/false);
        }

        const float b1v = b1[n];
        const float th  = thr[n];
        const float w2v = W2[n];

        // ---- 50-step synaptic recurrence, 8 independent elements/lane ----
        float x1v[8], syn[8], mem[8], ss[8];
        #pragma unroll
        for (int i = 0; i < 8; ++i) {
            x1v[i] = c[i] + b1v;
            syn[i] = 0.0f; mem[i] = 0.0f; ss[i] = 0.0f;
        }
        #pragma unroll 1
        for (int step = 0; step < NUM_STEPS; ++step) {
            #pragma unroll
            for (int i = 0; i < 8; ++i) {
                const float reset = (mem[i] > th) ? th : 0.0f;      // reset*thr (pre-update mem)
                syn[i] = SNN_ALPHA * syn[i] + x1v[i];
                mem[i] = beta_c * mem[i] + syn[i] - reset;
                ss[i] += (mem[i] > th) ? 1.0f : 0.0f;               // heaviside(mem - thr)
            }
        }
        #pragma unroll
        for (int i = 0; i < 8; ++i) acc[i] += ss[i] * w2v;          // fold GEMV weight
    }

    // C/D layout: vgpr i -> row M = i + 8*laneHi. Reduce across N (lanes/waves).
    #pragma unroll
    for (int i = 0; i < 8; ++i)
        atomicAdd(&rowsum[i + 8 * laneHi], acc[i]);                 // ds_add_f32

    __syncthreads();
    if (tid < ROWS_PER_BLOCK)
        out[m0 + tid] = rowsum[tid] + b2[0];
}

extern "C" void kernel_launch(void* const* d_in, const int* in_sizes, int n_in,
                              void* d_out, int out_size, void* d_ws, size_t ws_size,
                              hipStream_t stream) {
    const float* x    = (const float*)d_in[0];
    const float* W1   = (const float*)d_in[1];
    const float* b1   = (const float*)d_in[2];
    const float* thr  = (const float*)d_in[3];
    const float* beta = (const float*)d_in[4];
    const float* W2   = (const float*)d_in[5];
    const float* b2   = (const float*)d_in[6];
    float* out = (float*)d_out;

    const int B = in_sizes[0] / NIN;               // 2048
    dim3 grid(B / ROWS_PER_BLOCK), block(256);
    hipLaunchKernelGGL(snn_fused_kernel, grid, block, 0, stream,
                       x, W1, b1, thr, beta, W2, b2, out);
}